// YoloLayer_5669356832639
// MI455X (gfx1250) — compile-verified
//
#include <hip/hip_runtime.h>
#include <hip/hip_bf16.h>

typedef __attribute__((ext_vector_type(2))) float v2f;
typedef __attribute__((ext_vector_type(8))) float v8f;

// ---------------------------------------------------------------------------
// Exact-f32 wave32 sum via V_WMMA_F32_16X16X4_F32:
//   A(16x4): lane L<16 -> A[L][0]=a[0], A[L][1]=a[1]; lane L>=16 -> A[L-16][2],[3].
//   With a = {g, 0}, B = ones:  D[m][n] = g_m + g_{16+m}.
//   D lane l, vgpr r: M = r + 8*(l>=16).  Sum of d[0..7] gives half-wave sums;
//   one shfl_xor(16) completes the full 32-lane sum (all f32 adds, exact).
// ---------------------------------------------------------------------------
__device__ __forceinline__ float wave_sum32(float g) {
#if defined(__gfx1250__) && __has_builtin(__builtin_amdgcn_wmma_f32_16x16x4_f32)
    v2f a; a[0] = g;    a[1] = 0.0f;
    v2f b; b[0] = 1.0f; b[1] = 1.0f;
    v8f c = {};
    v8f d = __builtin_amdgcn_wmma_f32_16x16x4_f32(
        /*neg_a=*/false, a, /*neg_b=*/false, b,
        /*c_mod=*/(short)0, c, /*reuse_a=*/false, /*reuse_b=*/false);
    float s = d[0] + d[1] + d[2] + d[3] + d[4] + d[5] + d[6] + d[7];
    s += __shfl_xor(s, 16, 32);
    return s;
#else
    float s = g;
    for (int off = 16; off > 0; off >>= 1) s += __shfl_xor(s, off, 32);
    return s;
#endif
}

// corners: c = cos(atan2(im,re)) = re/|v|, s = im/|v|
__device__ __forceinline__ void box_corners(float x, float y, float w, float l,
                                            float im, float re,
                                            float* cx, float* cy) {
    float inv = 1.0f / sqrtf(im * im + re * re);
    float c = re * inv, s = im * inv;
    float hw = 0.5f * w, hl = 0.5f * l;
    cx[0] = x - hw * c - hl * s;  cy[0] = y - hw * s + hl * c;
    cx[1] = x - hw * c + hl * s;  cy[1] = y - hw * s - hl * c;
    cx[2] = x + hw * c + hl * s;  cy[2] = y + hw * s - hl * c;
    cx[3] = x + hw * c - hl * s;  cy[3] = y + hw * s + hl * c;
}

// Sutherland-Hodgman clip of subject quad (sx,sy) by clip quad (tcx,tcy),
// replicating the reference's semantics exactly (freeze keeps pre-clip poly).
__device__ __forceinline__ float clip_area(const float* pcx, const float* pcy,
                                           const float* tcx, const float* tcy) {
    float sx[8] = {}, sy[8] = {};
    for (int i = 0; i < 4; ++i) { sx[i] = pcx[i]; sy[i] = pcy[i]; }
    int m = 4;
    bool frozen = false;

    #pragma unroll
    for (int e = 0; e < 4; ++e) {
        float pxe = tcx[e],           pye = tcy[e];
        float qxe = tcx[(e + 1) & 3], qye = tcy[(e + 1) & 3];
        float a  = qye - pye;
        float b  = pxe - qxe;
        float cc = qxe * pye - qye * pxe;
        if (!frozen && m > 2) {
            float vals[8];
            for (int i = 0; i < 8; ++i) vals[i] = a * sx[i] + b * sy[i] + cc;
            float nx2[8] = {}, ny2[8] = {};
            int nm = 0;
            for (int i = 0; i < 8; ++i) {
                if (i < m) {
                    int j = (i + 1 >= m) ? 0 : i + 1;
                    float vi = vals[i], vj = vals[j];
                    if (vi <= 0.0f && nm < 8) {            // keep point i
                        nx2[nm] = sx[i]; ny2[nm] = sy[i]; ++nm;
                    }
                    if (vi * vj < 0.0f && nm < 8) {        // edge crosses line
                        float a2 = sy[j] - sy[i];
                        float b2 = sx[i] - sx[j];
                        float c2 = sx[j] * sy[i] - sy[j] * sx[i];
                        float wd = a * b2 - b * a2;
                        nx2[nm] = (b * c2 - cc * b2) / wd;
                        ny2[nm] = (cc * a2 - a * c2) / wd;
                        ++nm;
                    }
                }
            }
            if (nm > 0) {
                for (int i = 0; i < 8; ++i) { sx[i] = nx2[i]; sy[i] = ny2[i]; }
                m = nm;
            } else {
                frozen = true;   // reference quirk: keep pre-clip polygon
            }
        }
    }

    if (m > 2) {
        float acc = 0.0f;
        for (int i = 0; i < 8; ++i) {
            if (i < m) {
                int j = (i + 1 >= m) ? 0 : i + 1;
                acc += sx[i] * sy[j] - sy[i] * sx[j];
            }
        }
        return 0.5f * fabsf(acc);
    }
    return 0.0f;
}

// Literal replication of the reference O(8^3) hull-area formulation.
__device__ __forceinline__ float hull_area8(const float* hx, const float* hy) {
    float acc = 0.0f;
    for (int i = 0; i < 8; ++i) {
        for (int j = 0; j < 8; ++j) {
            float ex = hx[j] - hx[i];
            float ey = hy[j] - hy[i];
            bool left = true;
            #pragma unroll
            for (int k = 0; k < 8; ++k) {
                float cr = ex * (hy[k] - hy[i]) - ey * (hx[k] - hx[i]);
                left = left && (cr >= -1e-6f);
            }
            bool valid = left && (ex * ex + ey * ey > 1e-12f);
            acc += valid ? (hx[i] * hy[j] - hy[i] * hx[j]) : 0.0f;
        }
    }
    return 0.5f * fabsf(acc);
}

__global__ __launch_bounds__(256) void giou_kernel(const float* __restrict__ pred,
                                                   const float* __restrict__ targ,
                                                   float* __restrict__ iou_out,
                                                   float* __restrict__ partials,
                                                   int n) {
    int i   = blockIdx.x * 256 + threadIdx.x;
    int idx = (i < n) ? i : (n - 1);

    const float* pb = pred + (size_t)idx * 6;
    const float* tb = targ + (size_t)idx * 6;
    float pX = pb[0], pY = pb[1], pW = pb[2], pL = pb[3], pIm = pb[4], pRe = pb[5];
    float tX = tb[0], tY = tb[1], tW = tb[2], tL = tb[3], tIm = tb[4], tRe = tb[5];

    float pcx[4], pcy[4], tcx[4], tcy[4];
    box_corners(pX, pY, pW, pL, pIm, pRe, pcx, pcy);
    box_corners(tX, tY, tW, tL, tIm, tRe, tcx, tcy);

    float inter = clip_area(pcx, pcy, tcx, tcy);

    float hx[8], hy[8];
    for (int k = 0; k < 4; ++k) { hx[k] = pcx[k]; hy[k] = pcy[k]; }
    for (int k = 0; k < 4; ++k) { hx[4 + k] = tcx[k]; hy[4 + k] = tcy[k]; }
    float hull = hull_area8(hx, hy);

    float p_area = pW * pL;
    float t_area = tW * tL;
    float uni    = p_area + t_area - inter;
    float iou    = inter / (uni + 1e-16f);
    float giou   = 1.0f - (iou - (hull - uni) / (hull + 1e-16f));

    if (i < n) iou_out[i] = iou;
    float g = (i < n) ? giou : 0.0f;

    // wave32 sum via WMMA, then 8-wave LDS combine -> one partial per block
    float wsum = wave_sum32(g);
    __shared__ float lds[8];
    int lane = threadIdx.x & 31;
    int wave = threadIdx.x >> 5;
    if (lane == 0) lds[wave] = wsum;
    __syncthreads();
    if (threadIdx.x == 0) {
        float t = 0.0f;
        #pragma unroll
        for (int wv = 0; wv < 8; ++wv) t += lds[wv];
        partials[blockIdx.x] = t;
    }
}

__global__ __launch_bounds__(256) void reduce_partials(const float* __restrict__ partials,
                                                       int np, float* __restrict__ out) {
    __shared__ float sm[256];
    int t = threadIdx.x;
    float s = 0.0f;
    for (int i = t; i < np; i += 256) s += partials[i];
    sm[t] = s;
    __syncthreads();
    for (int k = 128; k > 0; k >>= 1) {
        if (t < k) sm[t] += sm[t + k];
        __syncthreads();
    }
    if (t == 0) out[0] = sm[0];
}

extern "C" void kernel_launch(void* const* d_in, const int* in_sizes, int n_in,
                              void* d_out, int out_size, void* d_ws, size_t ws_size,
                              hipStream_t stream) {
    const float* pred = (const float*)d_in[0];
    const float* targ = (const float*)d_in[1];
    float* out = (float*)d_out;
    int n  = in_sizes[0] / 6;          // (N,6) boxes
    int nb = (n + 255) / 256;
    float* partials = (float*)d_ws;    // nb floats of scratch

    giou_kernel<<<nb, 256, 0, stream>>>(pred, targ, out, partials, n);
    reduce_partials<<<1, 256, 0, stream>>>(partials, nb, out + n);
}